// ConvG_43026982371446
// MI455X (gfx1250) — compile-verified
//
#include <hip/hip_runtime.h>

// ---------------------------------------------------------------------------
// GCN forward: h = x @ W^T ; 2 hops of h <- A_hat @ h  (symmetric GCN norm)
//
// MI455X reasoning:
//  * GEMM: 1.6 GFLOP / ~51MB -> trivially compute-covered; use f32 WMMA
//    (V_WMMA_F32_16X16X4_F32), one wave per 16x16 tile, K=128 in 32 steps.
//  * Propagation: ~1.3 GB of gather + atomic traffic dominates (~60us at
//    23.3 TB/s). One wave per edge: lane j gathers float4 (512B/wave
//    coalesced), scales by norm, 4x global_atomic_add_f32 into the dest row.
//  * Self-loop term written as plain-store initialization of the hop output
//    (replaces a zeroing pass and removes N*128 atomics).
// ---------------------------------------------------------------------------

typedef __attribute__((ext_vector_type(2))) float v2f;
typedef __attribute__((ext_vector_type(8))) float v8f;

#define D 128            // D_IN == D_OUT == 128
#define NT_N (D / 16)    // 8 column tiles

// ---------------- degree / normalization ----------------------------------

__global__ void deg_init_kernel(float* deg, int n) {
    int i = blockIdx.x * blockDim.x + threadIdx.x;
    if (i < n) deg[i] = 1.0f;   // self-loop weight
}

__global__ void deg_acc_kernel(const int* __restrict__ col,
                               const float* __restrict__ w,
                               float* deg, int e) {
    int i = blockIdx.x * blockDim.x + threadIdx.x;
    if (i < e) atomicAdd(&deg[col[i]], w[i]);
}

__global__ void dinv_kernel(float* deg, int n) {   // in-place deg -> deg^{-1/2}
    int i = blockIdx.x * blockDim.x + threadIdx.x;
    if (i < n) {
        float d = deg[i];
        deg[i] = (d > 0.0f) ? rsqrtf(d) : 0.0f;
    }
}

// ---------------- dense GEMM via f32 WMMA ----------------------------------
// h[m][n] = sum_k x[m][k] * W[n][k]   (W is [out,in] row-major, so B = W^T)

__global__ void gemm_wmma_kernel(const float* __restrict__ x,
                                 const float* __restrict__ W,
                                 float* __restrict__ h, int n_tiles) {
    const int lane = threadIdx.x & 31;
    const int wave = threadIdx.x >> 5;
    const int tile = blockIdx.x * (blockDim.x >> 5) + wave;   // wave-uniform
    if (tile >= n_tiles) return;                              // whole-wave exit

    const int tm = tile / NT_N;      // 16-row block of nodes
    const int tn = tile % NT_N;      // 16-col block of outputs

    const int half = lane >> 4;      // 0: lanes 0-15, 1: lanes 16-31
    const int lm   = lane & 15;

    // A: lane holds row M=lm of x-tile, K slots {2*half, 2*half+1}
    const float* arow = x + (size_t)(tm * 16 + lm) * D;
    // B = W^T: lane holds col N=lm, i.e. row (tn*16+lm) of W, same K slots
    const float* brow = W + (size_t)(tn * 16 + lm) * D;

    v8f acc = {};
    #pragma unroll
    for (int k0 = 0; k0 < D; k0 += 4) {
        v2f a, b;
        a[0] = arow[k0 + 2 * half + 0];
        a[1] = arow[k0 + 2 * half + 1];
        b[0] = brow[k0 + 2 * half + 0];
        b[1] = brow[k0 + 2 * half + 1];
        acc = __builtin_amdgcn_wmma_f32_16x16x4_f32(
            /*neg_a=*/false, a, /*neg_b=*/false, b,
            /*c_mod=*/(short)0, acc, /*reuse_a=*/false, /*reuse_b=*/false);
    }

    // C/D layout: VGPR v -> M = v + 8*half, N = lm
    #pragma unroll
    for (int v = 0; v < 8; ++v) {
        int m = tm * 16 + v + 8 * half;
        h[(size_t)m * D + tn * 16 + lm] = acc[v];
    }
}

// ---------------- propagation hop ------------------------------------------

// hout[i][:] = dinv[i]^2 * hin[i][:]   (self-loop term, also initializes hout)
__global__ void self_init_kernel(const float* __restrict__ dinv,
                                 const float* __restrict__ hin,
                                 float* __restrict__ hout, int n) {
    int gid = blockIdx.x * blockDim.x + threadIdx.x;   // one float4 per thread
    if (gid >= n * (D / 4)) return;
    int node = gid >> 5;                               // 32 float4 per row
    float d = dinv[node];
    float4 v = ((const float4*)hin)[gid];
    float s = d * d;
    float4 o = {s * v.x, s * v.y, s * v.z, s * v.w};
    ((float4*)hout)[gid] = o;
}

// one wave per edge: hout[c][:] += dinv[r]*w*dinv[c] * hin[r][:]
__global__ void prop_edges_kernel(const int* __restrict__ row,
                                  const int* __restrict__ col,
                                  const float* __restrict__ w,
                                  const float* __restrict__ dinv,
                                  const float* __restrict__ hin,
                                  float* __restrict__ hout, int e) {
    const int lane = threadIdx.x & 31;
    const int edge = (blockIdx.x * blockDim.x + threadIdx.x) >> 5;
    if (edge >= e) return;

    int r = 0, c = 0;
    float nw = 0.0f;
    if (lane == 0) {
        r = row[edge];
        c = col[edge];
        nw = dinv[r] * w[edge] * dinv[c];
    }
    r  = __shfl(r, 0);
    c  = __shfl(c, 0);
    nw = __shfl(nw, 0);

    float4 v = ((const float4*)(hin + (size_t)r * D))[lane];  // 512B per wave
    float* dst = hout + (size_t)c * D + lane * 4;
    atomicAdd(dst + 0, nw * v.x);
    atomicAdd(dst + 1, nw * v.y);
    atomicAdd(dst + 2, nw * v.z);
    atomicAdd(dst + 3, nw * v.w);
}

// ---------------- launcher --------------------------------------------------

extern "C" void kernel_launch(void* const* d_in, const int* in_sizes, int n_in,
                              void* d_out, int out_size, void* d_ws, size_t ws_size,
                              hipStream_t stream) {
    const float* x  = (const float*)d_in[0];        // [N, 128]
    const int*   ei = (const int*)d_in[1];          // [2, E]
    const float* ew = (const float*)d_in[2];        // [E]
    const float* W  = (const float*)d_in[3];        // [128, 128]

    const int N = in_sizes[0] / D;
    const int E = in_sizes[2];
    const int* row = ei;        // edge_index[0] = source
    const int* col = ei + E;    // edge_index[1] = destination

    // workspace: [0, N) dinv ; then hop buffer h1 [N*D] (256B-aligned offset)
    float* dinv = (float*)d_ws;
    float* h1   = (float*)d_ws + ((N + 63) / 64) * 64;
    float* h0   = (float*)d_out;                    // GEMM result lives in d_out

    const int TB = 256;

    // 1) symmetric GCN normalization
    deg_init_kernel<<<(N + TB - 1) / TB, TB, 0, stream>>>(dinv, N);
    deg_acc_kernel<<<(E + TB - 1) / TB, TB, 0, stream>>>(col, ew, dinv, E);
    dinv_kernel<<<(N + TB - 1) / TB, TB, 0, stream>>>(dinv, N);

    // 2) h0 = x @ W^T  (WMMA f32, 8 waves / block, one 16x16 tile per wave)
    const int n_tiles = (N / 16) * NT_N;            // N = 50000 is 16-divisible
    gemm_wmma_kernel<<<(n_tiles + 7) / 8, TB, 0, stream>>>(x, W, h0, n_tiles);

    // 3) hop 1: h1 = A_hat @ h0
    self_init_kernel<<<(N * (D / 4) + TB - 1) / TB, TB, 0, stream>>>(dinv, h0, h1, N);
    prop_edges_kernel<<<(E * 32 + TB - 1) / TB, TB, 0, stream>>>(row, col, ew, dinv, h0, h1, E);

    // 4) hop 2: d_out = A_hat @ h1
    self_init_kernel<<<(N * (D / 4) + TB - 1) / TB, TB, 0, stream>>>(dinv, h1, h0, N);
    prop_edges_kernel<<<(E * 32 + TB - 1) / TB, TB, 0, stream>>>(row, col, ew, dinv, h1, h0, E);
}